// CausalGraphDiscovery_74105365725290
// MI455X (gfx1250) — compile-verified
//
#include <hip/hip_runtime.h>
#include <hip/hip_bf16.h>
#include <math.h>

// ---------------------------------------------------------------------------
// CausalGraphDiscovery on gfx1250 (CDNA5, wave32, WMMA + async-to-LDS)
//
//   1) struct_kernel : avg->corr->structure MLP->adj, parent_mask, has_parents
//   2) mech_kernel   : per-variable mechanism MLPs via V_WMMA_F32_16X16X4_F32,
//                      weights staged with GLOBAL_LOAD_ASYNC_TO_LDS_B128
//   3) tau_kernel    : pairwise tau net, deterministic block reduction
// ---------------------------------------------------------------------------

typedef __attribute__((ext_vector_type(2))) float v2f;
typedef __attribute__((ext_vector_type(8))) float v8f;

__device__ __forceinline__ v8f wmma_f32_16x16x4(v2f a, v2f b, v8f c) {
  // 8 args: (neg_a, A, neg_b, B, c_mod, C, reuse_a, reuse_b)
  return __builtin_amdgcn_wmma_f32_16x16x4_f32(false, a, false, b, (short)0, c,
                                               false, false);
}

__device__ __forceinline__ float sigmoidf_dev(float x) {
  return 1.0f / (1.0f + expf(-x));
}

// Async DMA: 16 bytes global -> LDS, tracked by ASYNCcnt.
__device__ __forceinline__ void async_g2l_b128(uint32_t lds_off, uint64_t gaddr) {
  asm volatile("global_load_async_to_lds_b128 %0, %1, off"
               :: "v"(lds_off), "v"(gaddr) : "memory");
}
__device__ __forceinline__ void wait_asynccnt0() {
  asm volatile("s_wait_asynccnt 0x0" ::: "memory");
}

// ---------------------------------------------------------------------------
// Kernel 1: single block, 256 threads. Tiny sequential part.
// ---------------------------------------------------------------------------
__global__ __launch_bounds__(256) void struct_kernel(
    const float* __restrict__ data,   // (16,512,16)
    const float* __restrict__ Ws1, const float* __restrict__ bs1,
    const float* __restrict__ Ws2, const float* __restrict__ bs2,
    const float* __restrict__ Ws3, const float* __restrict__ bs3,
    float* __restrict__ adj_out,      // d_out[0:256]
    float* __restrict__ scores_out,   // d_out[256+131072 : +256] (zeroed here)
    float* __restrict__ pm_out,       // d_ws[0:240]
    float* __restrict__ hp_out)       // d_ws[240:256]
{
  __shared__ float avg[8192];   // 512 x 16, 32 KB
  __shared__ float cm[16];
  __shared__ float covm[256];
  __shared__ float corr[256];
  __shared__ float h1[256];
  __shared__ float h2[128];
  __shared__ float adjS[256];

  const int t = threadIdx.x;

  for (int e = t; e < 8192; e += 256) {
    float s = 0.f;
    for (int b = 0; b < 16; ++b) s += data[b * 8192 + e];
    avg[e] = s * (1.0f / 16.0f);
  }
  __syncthreads();

  if (t < 16) {
    float s = 0.f;
    for (int q = 0; q < 512; ++q) s += avg[q * 16 + t];
    cm[t] = s * (1.0f / 512.0f);
  }
  __syncthreads();

  {
    const int i = t >> 4, j = t & 15;
    const float mi = cm[i], mj = cm[j];
    float s = 0.f;
    for (int q = 0; q < 512; ++q)
      s += (avg[q * 16 + i] - mi) * (avg[q * 16 + j] - mj);
    covm[t] = s;
  }
  __syncthreads();

  {
    const int i = t >> 4, j = t & 15;
    const float di = sqrtf(fmaxf(covm[i * 17], 0.f));
    const float dj = sqrtf(fmaxf(covm[j * 17], 0.f));
    const float den = di * dj;
    float c = (den > 0.f) ? fabsf(covm[t] / den) : 0.f;
    if (i == j) c = 0.f;
    corr[t] = c;
  }
  __syncthreads();

  {
    float s = bs1[t];
    for (int k = 0; k < 256; ++k) s += corr[k] * Ws1[k * 256 + t];
    h1[t] = fmaxf(s, 0.f);
  }
  __syncthreads();

  if (t < 128) {
    float s = bs2[t];
    for (int k = 0; k < 256; ++k) s += h1[k] * Ws2[k * 128 + t];
    h2[t] = fmaxf(s, 0.f);
  }
  __syncthreads();

  {
    float s = bs3[t];
    for (int k = 0; k < 128; ++k) s += h2[k] * Ws3[k * 256 + t];
    float a = sigmoidf_dev(s);
    const int i = t >> 4, j = t & 15;
    if (j <= i) a = 0.f;
    adjS[t] = a;
    adj_out[t] = a;
    scores_out[t] = 0.f;   // zero scores region (diagonal stays 0)
  }
  __syncthreads();

  if (t < 240) {
    const int v = t / 15, p = t % 15;
    const int parent = (p < v) ? p : p + 1;
    pm_out[t] = (adjS[parent * 16 + v] > 0.5f) ? 1.f : 0.f;
  }
  __syncthreads();

  if (t < 16) {
    float any = 0.f;
    for (int p = 0; p < 15; ++p) any += pm_out[t * 15 + p];
    hp_out[t] = (any > 0.f) ? 1.f : 0.f;
  }
}

// ---------------------------------------------------------------------------
// Kernel 2: mechanism MLPs via fp32 WMMA. 2 waves / block, 1 M-tile / wave.
// grid = 256 blocks -> 512 M-tiles = 8192 rows. Loops over 16 variables.
// LDS: sB2 32KB + sB1 8KB + 2x sC1 8KB = 56KB
// ---------------------------------------------------------------------------
#define MECH_WAVES 2

__global__ __launch_bounds__(64) void mech_kernel(
    const float* __restrict__ data,  // flat: 8192 x 16
    const float* __restrict__ Wm1,   // 16 x 15 x 128
    const float* __restrict__ bm1,   // 16 x 128
    const float* __restrict__ Wm2,   // 16 x 128 x 64
    const float* __restrict__ bm2,   // 16 x 64
    const float* __restrict__ Wm3,   // 16 x 64
    const float* __restrict__ bm3,   // 16
    const float* __restrict__ pm,    // 16 x 15 (0/1)
    const float* __restrict__ hp,    // 16 (0/1)
    float* __restrict__ pred)        // 8192 x 16
{
  __shared__ __align__(16) float sB2[128 * 64];              // Wm2[v], 32 KB
  __shared__ __align__(16) float sB1[16 * 128];              // B1' (gathered), 8 KB
  __shared__ __align__(16) float sC1[MECH_WAVES][16 * 128];  // 16 KB

  const int tid  = threadIdx.x;
  const int wave = tid >> 5;
  const int lane = tid & 31;
  const int half = lane >> 4;   // 0: lanes 0-15, 1: lanes 16-31
  const int lm   = lane & 15;
  const int mtile   = blockIdx.x * MECH_WAVES + wave;  // 0..511, exact
  const int rowbase = mtile * 16;

  // A operand (flat rows) in documented 32-bit A layout:
  //   lane m (0-15): VGPR0=K(k0), VGPR1=K(k0+1);  lanes 16-31: K(k0+2),K(k0+3)
  v2f aflat[4];
#pragma unroll
  for (int ks = 0; ks < 4; ++ks) {
    const float* pA = data + (rowbase + lm) * 16 + ks * 4 + half * 2;
    v2f a; a.x = pA[0]; a.y = pA[1];
    aflat[ks] = a;
  }

  for (int v = 0; v < 16; ++v) {
    __syncthreads();  // protect sB1/sB2 readers of previous iteration

    // ---- stage Wm2[v] (128x64 f32 = 8192 B x4) via async DMA to LDS ------
    {
      const uint64_t gbase = (uint64_t)(uintptr_t)(Wm2 + v * 8192);
      const uint32_t lbase = (uint32_t)(uintptr_t)(&sB2[0]);
      for (int e = tid; e < 512; e += 64)      // 512 x 16B chunks
        async_g2l_b128(lbase + (uint32_t)(e * 16), gbase + (uint64_t)(e * 16));
    }

    // ---- stage B1'[j][h] = (j==v) ? 0 : pm[v][pj]*Wm1[v][pj][h] ----------
    // (branchless: clamped index + unconditional loads + cndmask)
    {
      const float* Wm1v = Wm1 + v * 15 * 128;
      for (int e = tid; e < 2048; e += 64) {
        const int j = e >> 7, h = e & 127;
        int pj = j - ((j > v) ? 1 : 0);
        pj = (j == v) ? 0 : pj;                   // stays in [0,14]
        const float pmv = pm[v * 15 + pj];        // unconditional load
        const float wv  = Wm1v[pj * 128 + h];     // unconditional load
        sB1[e] = (j == v) ? 0.f : pmv * wv;
      }
    }
    wait_asynccnt0();   // per-wave async copies complete
    __syncthreads();    // all waves' staging visible

    // ---------------- GEMM1: [16x16(flat)] x [16x128(B1')] ----------------
    v8f acc1[8] = {};
#pragma unroll
    for (int ks = 0; ks < 4; ++ks) {
      const int j0 = ks * 4 + half * 2;
#pragma unroll
      for (int nt = 0; nt < 8; ++nt) {
        const int n = nt * 16 + lm;
        v2f b; b.x = sB1[j0 * 128 + n]; b.y = sB1[(j0 + 1) * 128 + n];
        acc1[nt] = wmma_f32_16x16x4(aflat[ks], b, acc1[nt]);
      }
    }

    // bias + relu -> per-wave LDS buffer (row-major 16 x 128) for re-layout
    float* myC1 = &sC1[wave][0];
#pragma unroll
    for (int nt = 0; nt < 8; ++nt) {
      const int n = nt * 16 + lm;
      const float bias = bm1[v * 128 + n];
#pragma unroll
      for (int g = 0; g < 8; ++g) {
        const int m = g + half * 8;  // D layout: VGPR g -> M=g / M=g+8
        myC1[m * 128 + n] = fmaxf(acc1[nt][g] + bias, 0.f);
      }
    }
    // same-wave LDS ops are in-order: stores above precede loads below

    // ---------------- GEMM2: [16x128(C1)] x [128x64(Wm2)] -----------------
    v8f acc2[4] = {};
#pragma unroll 8
    for (int ks = 0; ks < 32; ++ks) {
      const int k0 = ks * 4 + half * 2;
      const v2f a = *reinterpret_cast<const v2f*>(&myC1[lm * 128 + k0]);
#pragma unroll
      for (int nt = 0; nt < 4; ++nt) {
        const int n = nt * 16 + lm;
        v2f b; b.x = sB2[k0 * 64 + n]; b.y = sB2[(k0 + 1) * 64 + n];
        acc2[nt] = wmma_f32_16x16x4(a, b, acc2[nt]);
      }
    }

    // ------------- layer-2 bias+relu, dot with Wm3 (GEMM3, N=1) -----------
    float part[8];
#pragma unroll
    for (int g = 0; g < 8; ++g) part[g] = 0.f;
#pragma unroll
    for (int nt = 0; nt < 4; ++nt) {
      const int n = nt * 16 + lm;
      const float b2 = bm2[v * 64 + n];
      const float w3 = Wm3[v * 64 + n];
#pragma unroll
      for (int g = 0; g < 8; ++g)
        part[g] += fmaxf(acc2[nt][g] + b2, 0.f) * w3;
    }
    // butterfly-reduce across the 16 lanes of each half (masks 1,2,4,8)
#pragma unroll
    for (int g = 0; g < 8; ++g) {
#pragma unroll
      for (int off = 1; off < 16; off <<= 1)
        part[g] += __shfl_xor(part[g], off, 32);
    }
    // select this lane's row value: row = rowbase + half*8 + lm (lm < 8)
    float mech = part[0];
#pragma unroll
    for (int g = 1; g < 8; ++g) mech = (lm == g) ? part[g] : mech;

    const float hpv = hp[v];
    const float b3  = bm3[v];
    if (lm < 8) {
      const int row = rowbase + half * 8 + lm;
      const float fallback = data[row * 16 + v];
      pred[row * 16 + v] = (hpv > 0.f) ? (mech + b3) : fallback;
    }
  }
}

// ---------------------------------------------------------------------------
// Kernel 3: pairwise tau net. One block per pair; deterministic reduction.
// ---------------------------------------------------------------------------
__global__ __launch_bounds__(256) void tau_kernel(
    const float* __restrict__ data,  // flat: 8192 x 16
    const float* __restrict__ Wt1,   // 240 x 2 x 32
    const float* __restrict__ bt1,   // 240 x 32
    const float* __restrict__ Wt2,   // 240 x 32
    const float* __restrict__ bt2,   // 240
    float* __restrict__ scores)      // 16 x 16 (off-diagonal written)
{
  const int p = blockIdx.x;               // 0..239
  const int i = p / 15;
  const int jj = p % 15;
  const int j = (jj < i) ? jj : jj + 1;

  const float* w1a = Wt1 + p * 64;        // Wt1[p][0][:]
  const float* w1b = w1a + 32;            // Wt1[p][1][:]
  const float* b1  = bt1 + p * 32;
  const float* w2  = Wt2 + p * 32;
  const float bb   = bt2[p];

  float acc = 0.f;
  for (int n = threadIdx.x; n < 8192; n += 256) {
    const float x0 = data[n * 16 + i];
    const float x1 = data[n * 16 + j];
    float s = bb;
#pragma unroll
    for (int h = 0; h < 32; ++h) {
      const float ht = fmaxf(x0 * w1a[h] + x1 * w1b[h] + b1[h], 0.f);
      s += ht * w2[h];
    }
    acc += sigmoidf_dev(s);
  }

  __shared__ float red[256];
  red[threadIdx.x] = acc;
  __syncthreads();
  for (int st = 128; st > 0; st >>= 1) {
    if (threadIdx.x < st) red[threadIdx.x] += red[threadIdx.x + st];
    __syncthreads();
  }
  if (threadIdx.x == 0) scores[i * 16 + j] = red[0] * (1.0f / 8192.0f);
}

// ---------------------------------------------------------------------------
extern "C" void kernel_launch(void* const* d_in, const int* in_sizes, int n_in,
                              void* d_out, int out_size, void* d_ws, size_t ws_size,
                              hipStream_t stream) {
  (void)in_sizes; (void)n_in; (void)out_size; (void)ws_size;
  const float* data = (const float*)d_in[0];
  const float* Ws1  = (const float*)d_in[1];
  const float* bs1  = (const float*)d_in[2];
  const float* Ws2  = (const float*)d_in[3];
  const float* bs2  = (const float*)d_in[4];
  const float* Ws3  = (const float*)d_in[5];
  const float* bs3  = (const float*)d_in[6];
  const float* Wm1  = (const float*)d_in[7];
  const float* bm1  = (const float*)d_in[8];
  const float* Wm2  = (const float*)d_in[9];
  const float* bm2  = (const float*)d_in[10];
  const float* Wm3  = (const float*)d_in[11];
  const float* bm3  = (const float*)d_in[12];
  const float* Wt1  = (const float*)d_in[13];
  const float* bt1  = (const float*)d_in[14];
  const float* Wt2  = (const float*)d_in[15];
  const float* bt2  = (const float*)d_in[16];

  float* out        = (float*)d_out;
  float* adj_out    = out;                  // 256
  float* pred_out   = out + 256;            // 131072
  float* scores_out = out + 256 + 131072;   // 256

  float* pm = (float*)d_ws;                 // 240 floats
  float* hp = pm + 240;                     // 16 floats

  struct_kernel<<<1, 256, 0, stream>>>(data, Ws1, bs1, Ws2, bs2, Ws3, bs3,
                                       adj_out, scores_out, pm, hp);
  mech_kernel<<<256, 64, 0, stream>>>(data, Wm1, bm1, Wm2, bm2, Wm3, bm3,
                                      pm, hp, pred_out);
  tau_kernel<<<240, 256, 0, stream>>>(data, Wt1, bt1, Wt2, bt2, scores_out);
}